// MendGraph_65326452572488
// MI455X (gfx1250) — compile-verified
//
#include <hip/hip_runtime.h>
#include <hip/hip_bf16.h>
#include <stdint.h>

// ---------------- problem constants (from reference) ----------------
#define N_NODES   100000
#define NUM_PRED  5
#define FEAT      128
#define NE        1600000
#define CAP       (N_NODES * NUM_PRED)            // 500000 new-node slots
#define X_ROWS    (N_NODES * (NUM_PRED + 1))      // 600000 feature rows
#define X_ELEMS   ((size_t)X_ROWS * FEAT)         // 76,800,000 f32
#define EO_COLS   (NE + 2 * CAP)                  // 2,600,000 per edge row
#define SCAN_BS   256
#define NB        ((N_NODES + SCAN_BS - 1) / SCAN_BS)   // 391 blocks

// clang ext-vector types (accepted by nontemporal builtins, unlike HIP float4)
typedef float v4f __attribute__((ext_vector_type(4)));
typedef int   v4i __attribute__((ext_vector_type(4)));

__device__ __forceinline__ v4f ntload4f(const float* p) {
    return __builtin_nontemporal_load((const v4f*)p);
}
__device__ __forceinline__ void ntstore4f(float* p, v4f v) {
    __builtin_nontemporal_store(v, (v4f*)p);
}
__device__ __forceinline__ v4i ntload4i(const int* p) {
    return __builtin_nontemporal_load((const v4i*)p);
}
__device__ __forceinline__ void ntstore4i(int* p, v4i v) {
    __builtin_nontemporal_store(v, (v4i*)p);
}

// ---------------- gfx1250 async LDS copy path (probe-guarded) -------
#if defined(__AMDGCN__) && \
    __has_builtin(__builtin_amdgcn_global_load_async_to_lds_b128) && \
    __has_builtin(__builtin_amdgcn_global_store_async_from_lds_b128) && \
    __has_builtin(__builtin_amdgcn_s_wait_asynccnt)
#define USE_ASYNC_LDS 1
typedef __attribute__((address_space(1))) v4i gv4_t;  // global int4-vec
typedef __attribute__((address_space(3))) v4i lv4_t;  // LDS int4-vec
#else
#define USE_ASYNC_LDS 0
#endif

__device__ __forceinline__ int pred_k(float pm) {
    // jnp.round == round half to even -> rintf (default RNE), then clip [0,5]
    float r = rintf(pm);
    r = fminf(fmaxf(r, 0.0f), (float)NUM_PRED);
    return (int)r;
}

// 1) per-block exclusive scan of k; write local offsets + block sums
__global__ void k_scan_block(const float* __restrict__ pm,
                             int* __restrict__ loc, int* __restrict__ bsum) {
    __shared__ int s[SCAN_BS];
    const int t = threadIdx.x;
    const int i = blockIdx.x * SCAN_BS + t;
    int k = 0;
    if (i < N_NODES) k = pred_k(pm[i]);
    s[t] = k;
    __syncthreads();
    for (int d = 1; d < SCAN_BS; d <<= 1) {
        int v = (t >= d) ? s[t - d] : 0;
        __syncthreads();
        s[t] += v;
        __syncthreads();
    }
    if (i < N_NODES) loc[i] = s[t] - k;         // exclusive within block
    if (t == SCAN_BS - 1) bsum[blockIdx.x] = s[t];
}

// 2) single-block exclusive scan of the NB block sums; emit total_new
__global__ void k_scan_sums(const int* __restrict__ bsum,
                            int* __restrict__ bpref, int* __restrict__ total) {
    __shared__ int s[512];
    const int t = threadIdx.x;
    int v = (t < NB) ? bsum[t] : 0;
    s[t] = v;
    __syncthreads();
    for (int d = 1; d < 512; d <<= 1) {
        int x = (t >= d) ? s[t - d] : 0;
        __syncthreads();
        s[t] += x;
        __syncthreads();
    }
    if (t < NB) bpref[t] = s[t] - v;            // exclusive
    if (t == 511) *total = s[511];              // total_new
}

// 3) bulk copy org_feats -> x_out[0:N] ; staged via async-LDS b128 on gfx1250
__global__ void k_copy_feats(const float* __restrict__ src, float* __restrict__ dst) {
    const size_t idx = (size_t)blockIdx.x * blockDim.x + threadIdx.x;  // one float4 each
#if USE_ASYNC_LDS
    __shared__ __align__(16) char stage[256 * 16];
    lv4_t* lp = (lv4_t*)(stage + threadIdx.x * 16);
    __builtin_amdgcn_global_load_async_to_lds_b128(
        (gv4_t*)((const v4i*)src + idx), lp, 0, 0);
    __builtin_amdgcn_s_wait_asynccnt(0);
    __builtin_amdgcn_global_store_async_from_lds_b128(
        (gv4_t*)((v4i*)dst + idx), lp, 0, 0);
    __builtin_amdgcn_s_wait_asynccnt(0);
#else
    ntstore4f(dst + idx * 4, ntload4f(src + idx * 4));
#endif
}

// 4) copy org_edges[2][NE] into the strided output rows (int4, non-temporal)
__global__ void k_copy_edges(const int* __restrict__ src, int* __restrict__ dst) {
    const int idx = blockIdx.x * blockDim.x + threadIdx.x;   // 2*NE/4 total
    const int per_row = NE / 4;
    const int row = idx / per_row;
    const int col = idx - row * per_row;
    v4i v = ntload4i(src + (size_t)row * NE + (size_t)col * 4);
    ntstore4i(dst + (size_t)row * EO_COLS + (size_t)col * 4, v);
}

// 5) one wave32 per node: scatter k valid gen rows + write edge pairs
__global__ void k_scatter(const float* __restrict__ pm, const float* __restrict__ gen,
                          const int* __restrict__ loc, const int* __restrict__ bpref,
                          float* __restrict__ xout, int* __restrict__ edges) {
    const int wave = (blockIdx.x * blockDim.x + threadIdx.x) >> 5;
    const int lane = threadIdx.x & 31;
    if (wave >= N_NODES) return;
    const int i = wave;
    const int k = pred_k(pm[i]);
    if (k == 0) return;
    const int off = loc[i] + bpref[i >> 8];     // global exclusive offset
    const float* src = gen + (size_t)i * (NUM_PRED * FEAT);
    __builtin_prefetch(src, 0, 1);              // global_prefetch of this node's rows
#pragma unroll
    for (int j = 0; j < NUM_PRED; ++j) {
        if (j >= k) break;
        const int s = off + j;
        v4f v = ntload4f(src + j * FEAT + lane * 4);               // read-once stream
        ntstore4f(xout + (size_t)(N_NODES + s) * FEAT + lane * 4, v);
        if (lane == 0) {
            const int nid = N_NODES + s;
            edges[NE + 2 * s]               = i;    // row0: (owner, new_id)
            edges[NE + 2 * s + 1]           = nid;
            edges[EO_COLS + NE + 2 * s]     = nid;  // row1: (new_id, owner)
            edges[EO_COLS + NE + 2 * s + 1] = i;
        }
    }
}

// 6) one wave32 per slot: zero rows + (-1) edges for slots >= total_new
__global__ void k_fill_invalid(const int* __restrict__ total,
                               float* __restrict__ xout, int* __restrict__ edges) {
    const int wave = (blockIdx.x * blockDim.x + threadIdx.x) >> 5;
    const int lane = threadIdx.x & 31;
    if (wave >= CAP) return;
    const int tn = *total;
    if (wave < tn) return;                       // valid slots written by k_scatter
    const int s = wave;
    v4f z = {0.f, 0.f, 0.f, 0.f};
    ntstore4f(xout + (size_t)(N_NODES + s) * FEAT + lane * 4, z);
    if (lane == 0) {
        edges[NE + 2 * s]               = -1;
        edges[NE + 2 * s + 1]           = -1;
        edges[EO_COLS + NE + 2 * s]     = -1;
        edges[EO_COLS + NE + 2 * s + 1] = -1;
    }
}

extern "C" void kernel_launch(void* const* d_in, const int* in_sizes, int n_in,
                              void* d_out, int out_size, void* d_ws, size_t ws_size,
                              hipStream_t stream) {
    (void)in_sizes; (void)n_in; (void)out_size; (void)ws_size;
    const float* org_feats    = (const float*)d_in[0];   // [N,128] f32
    const int*   org_edges    = (const int*)  d_in[1];   // [2,E]  i32
    const float* pred_missing = (const float*)d_in[2];   // [N]    f32
    const float* gen_feats    = (const float*)d_in[3];   // [N,640] f32

    float* xout  = (float*)d_out;                        // [600000,128] f32
    int*   edges = (int*)((float*)d_out + X_ELEMS);      // [2, 2.6M]   i32 (bit-exact)

    int* ws    = (int*)d_ws;
    int* loc   = ws;                 // N      local exclusive offsets
    int* bsum  = loc + N_NODES;      // NB     block sums
    int* bpref = bsum + NB;          // NB     block prefixes
    int* total = bpref + NB;         // 1      total_new

    // 1-2) prefix scan of k
    k_scan_block<<<NB, SCAN_BS, 0, stream>>>(pred_missing, loc, bsum);
    k_scan_sums<<<1, 512, 0, stream>>>(bsum, bpref, total);

    // 3) org_feats copy: N*FEAT/4 = 3,200,000 float4 -> 12,500 blocks of 256
    k_copy_feats<<<(N_NODES * FEAT / 4) / 256, 256, 0, stream>>>(org_feats, xout);

    // 4) org_edges copy: 2*NE/4 = 800,000 int4 -> 3125 blocks of 256
    k_copy_edges<<<(2 * NE / 4) / 256, 256, 0, stream>>>(org_edges, edges);

    // 5) compacted scatter: one wave per node, 8 waves per 256-thread block
    k_scatter<<<(N_NODES + 7) / 8, 256, 0, stream>>>(pred_missing, gen_feats,
                                                     loc, bpref, xout, edges);

    // 6) invalid-slot fill: one wave per slot
    k_fill_invalid<<<(CAP + 7) / 8, 256, 0, stream>>>(total, xout, edges);
}